// BaseModel_1288490189198
// MI455X (gfx1250) — compile-verified
//
#include <hip/hip_runtime.h>
#include <cstdint>

// Embedding gather + mean-pool (DLRM-style), tuned for MI455X / gfx1250.
//
// Pure random-gather workload: ~210 MB of 256 B embedding-row reads per call,
// tables L2-resident (4 x 25.6 MB hist tables < 192 MB L2). No matmul exists
// in this op, so WMMA does not apply; the CDNA5 feature that matters is the
// async global->LDS DMA path (ASYNCcnt): one wave keeps 25 gather transfers
// in flight with zero destination-VGPR pressure.
//
// Round-2 change: indices are preloaded into registers BEFORE the first
// asm-volatile barrier so the async issue loop is not serialized by
// per-iteration s_wait_loadcnt 0; async ops use the SADDR (base+offset32)
// form so per-row address math is a single 32-bit VALU op.

#define SEQ 50
#define DIM 64          // floats per embedding row (256 bytes)
#define OUTROW 512      // 8 features * 64

struct Params {
    const int*   hist[4];   // [B, SEQ] int32
    const int*   cat[4];    // [B, 1]   int32
    const float* Wh[4];     // [V, DIM] f32
    const float* Wc[4];     // [V, DIM] f32
    float*       out;       // [B, OUTROW] f32
};

// One wave32 per (batch b, feature f).
__global__ __launch_bounds__(32) void embed_pool_kernel(Params p) {
    const int b = blockIdx.x;
    const int f = blockIdx.y;
    const int t = threadIdx.x;               // 0..31

    __shared__ float smem[SEQ * DIM];        // 12.8 KB staging, private to this wave

    const int*   ids = p.hist[f] + (size_t)b * SEQ;
    const float* Wh  = p.Wh[f];

    // Low 32 bits of a generic LDS pointer == byte offset in the wave's LDS
    // allocation (flat->LDS aperture truncation) — the address space the
    // async-load VDST operand uses.
    const unsigned lds_base = (unsigned)(uintptr_t)&smem[0];

    const int half = t >> 4;                 // which of the 2 rows this lane serves
    const int quad = t & 15;                 // 16-byte chunk within the row

    const unsigned lane_goff = (unsigned)quad << 4;                        // bytes into row
    const unsigned lane_lo   = lds_base + ((unsigned)half << 8) + lane_goff;

    // 1) Preload all per-lane row indices first: 25 back-to-back dword loads
    //    (whole wave touches ids[0..49] = 200 B), no asm barrier in between.
    int idx[SEQ / 2];
    #pragma unroll
    for (int i = 0; i < SEQ / 2; ++i)
        idx[i] = ids[2 * i + half];

    // 2) Stream out all 25 async copies (2 rows x 256 B per instruction,
    //    16 B/lane). SADDR form: addr = s[Wh] + (id*256 + quad*16).
    //    Max offset 100000*256 ≈ 25.6 MB < 2^31.
    #pragma unroll
    for (int i = 0; i < SEQ / 2; ++i) {
        const unsigned goff = ((unsigned)idx[i] << 8) + lane_goff;
        const unsigned lo   = lane_lo + (unsigned)(i * 512);
        asm volatile("global_load_async_to_lds_b128 %0, %1, %2"
                     :: "v"(lo), "v"(goff), "s"(Wh)
                     : "memory");
    }

    // 3) Independent work while the gather DMA is in flight: single-id cat row.
    const int    cid = p.cat[f][b];
    const float2 cv  = ((const float2*)(p.Wc[f] + ((size_t)(unsigned)cid << 6)))[t];

    float* orow = p.out + (size_t)b * OUTROW;
    ((float2*)(orow + 256 + f * DIM))[t] = cv;

    // 4) Wait for this wave's async LDS writes; single-wave workgroup -> no barrier.
    asm volatile("s_wait_asynccnt 0" ::: "memory");

    // 5) Mean-pool: lane t owns columns d = 2t, 2t+1. Conflict-free LDS reads
    //    (32 lanes x 8 B = 256 B = one full 64-bank pass per row).
    float accx = 0.f, accy = 0.f;
    const float2* s2 = (const float2*)smem;
    #pragma unroll
    for (int r = 0; r < SEQ; ++r) {
        const float2 v = s2[r * (DIM / 2) + t];
        accx += v.x;
        accy += v.y;
    }
    const float inv = 1.0f / (float)SEQ;
    ((float2*)(orow + f * DIM))[t] = make_float2(accx * inv, accy * inv);
}

extern "C" void kernel_launch(void* const* d_in, const int* in_sizes, int n_in,
                              void* d_out, int out_size, void* d_ws, size_t ws_size,
                              hipStream_t stream) {
    (void)n_in; (void)d_ws; (void)ws_size; (void)out_size;

    Params p;
    // setup_inputs() insertion order: hist_i, cat_i, W_hist_i, W_cat_i for i=0..3
    for (int i = 0; i < 4; ++i) {
        p.hist[i] = (const int*)  d_in[4 * i + 0];
        p.cat[i]  = (const int*)  d_in[4 * i + 1];
        p.Wh[i]   = (const float*)d_in[4 * i + 2];
        p.Wc[i]   = (const float*)d_in[4 * i + 3];
    }
    p.out = (float*)d_out;

    const int B = in_sizes[1];               // cat0 has B elements

    dim3 grid((unsigned)B, 4, 1);
    embed_pool_kernel<<<grid, dim3(32, 1, 1), 0, stream>>>(p);
}